// aspect_rating_3_39900246180590
// MI455X (gfx1250) — compile-verified
//
#include <hip/hip_runtime.h>
#include <math.h>

#define R_ROWS 20000
#define L_WORDS 30
#define D_DIM 200
#define A_DIM 32
#define B_BATCH 4096
#define NNZ_CNT 65536
#define NUM_NEG 2

typedef __attribute__((ext_vector_type(2))) float v2f;
typedef __attribute__((ext_vector_type(8))) float v8f;

__device__ __forceinline__ float wave_sum(float v) {
#pragma unroll
  for (int m = 16; m >= 1; m >>= 1) v += __shfl_xor(v, m, 32);
  return v;
}
__device__ __forceinline__ float wave_max(float v) {
#pragma unroll
  for (int m = 16; m >= 1; m >>= 1) v = fmaxf(v, __shfl_xor(v, m, 32));
  return v;
}

__device__ __forceinline__ void wait_async0() {
#if __has_builtin(__builtin_amdgcn_s_wait_asynccnt)
  __builtin_amdgcn_s_wait_asynccnt(0);
#else
  asm volatile("s_wait_asynccnt 0" ::: "memory");
#endif
}

// ---------------------------------------------------------------------------
// Generic C[MxN] = A[MxK] @ B[KxN] (+bias[N]) using V_WMMA_F32_16X16X4_F32.
// A row-major (lda=K). B row-major (ldb=N) or, if bTrans, B is stored [N x K].
// One wave computes one 16x16 C tile; block = 8 waves sharing a B column tile
// staged in LDS. K must be a multiple of 4 (true for K=200 and K=32 here).
// ---------------------------------------------------------------------------
__global__ void __launch_bounds__(256)
gemm_wmma_f32(const float* __restrict__ A, const float* __restrict__ B,
              const float* __restrict__ bias, float* __restrict__ C,
              int M, int N, int K, int bTrans) {
  __shared__ float bT[200 * 16];  // max K = 200
  const int lane = threadIdx.x & 31;
  const int wave = threadIdx.x >> 5;
  const int nBase = blockIdx.x * 16;

  for (int idx = threadIdx.x; idx < K * 16; idx += 256) {
    int k = idx >> 4, n0 = idx & 15, gn = nBase + n0;
    float val = 0.f;
    if (gn < N) val = bTrans ? B[(size_t)gn * K + k] : B[(size_t)k * N + gn];
    bT[idx] = val;
  }
  __syncthreads();

  const int mtile = blockIdx.y * 8 + wave;
  if (mtile * 16 >= M) return;  // no further barriers below

  const int mrow  = mtile * 16 + (lane & 15);
  const int khalf = lane >> 4;  // lanes 16-31 hold K+2 / M+8 halves
  const float* arow = A + (size_t)mrow * K;

  v8f acc = {0.f, 0.f, 0.f, 0.f, 0.f, 0.f, 0.f, 0.f};
  for (int k0 = 0; k0 < K; k0 += 4) {
    const int kb = k0 + 2 * khalf;
    // A 16x4 frag: lane m=lane&15 holds K = kb, kb+1 (contiguous -> b64 load)
    v2f a = *reinterpret_cast<const v2f*>(arow + kb);
    // B 4x16 frag: lane n=lane&15 holds rows K = kb, kb+1 from LDS
    v2f b;
    b[0] = bT[kb * 16 + (lane & 15)];
    b[1] = bT[(kb + 1) * 16 + (lane & 15)];
    acc = __builtin_amdgcn_wmma_f32_16x16x4_f32(false, a, false, b,
                                                (short)0, acc, false, false);
  }

  const int n = nBase + (lane & 15);
  if (n < N) {
    const float bv = bias ? bias[n] : 0.f;
#pragma unroll
    for (int i = 0; i < 8; ++i) {
      const int m = mtile * 16 + i + 8 * khalf;  // C layout: VGPR i -> M=i / i+8
      C[(size_t)m * N + n] = acc[i] + bv;
    }
  }
}

// ---------------------------------------------------------------------------
// Fused ABAE attention: one block per review row r.
// dx[l] = e_w[r,l]·v[r] + M_b·y_s[r]; ax = softmax(dx); z_s = sum_l ax[l]*e_w[r,l]
// e_w row (24 KB) staged once in LDS via CDNA5 async global->LDS copies
// (GLOBAL_LOAD_ASYNC_TO_LDS_B128, ASYNCcnt) -> single streaming pass, no VGPR
// round-trip in the copy loop.
// ---------------------------------------------------------------------------
__global__ void __launch_bounds__(256)
attention_kernel(const float* __restrict__ e_w, const float* __restrict__ y_s,
                 const float* __restrict__ M_b, const float* __restrict__ v,
                 float* __restrict__ z_s) {
  __shared__ __align__(16) float ew[L_WORDS * D_DIM];
  __shared__ float vv[D_DIM];
  __shared__ float dxs[32];
  __shared__ float cvec;
  const int r = blockIdx.x;
  const int t = threadIdx.x;
  const int lane = t & 31, wave = t >> 5;

  // Async-stage the 24000B e_w row: 1500 x b128 lane transfers.
  const char* gbase = (const char*)(e_w + (size_t)r * (L_WORDS * D_DIM));
  const unsigned ewBase = (unsigned)(uintptr_t)ew;
  for (int i = t; i < (L_WORDS * D_DIM) / 4; i += 256) {
    unsigned la = ewBase + 16u * (unsigned)i;
    const char* ga = gbase + 16 * (size_t)i;
    asm volatile("global_load_async_to_lds_b128 %0, %1, off"
                 :: "v"(la), "v"(ga) : "memory");
  }
  for (int i = t; i < D_DIM; i += 256) vv[i] = v[(size_t)r * D_DIM + i];
  if (t == 0) cvec = 0.f;
  wait_async0();               // this wave's async copies landed in LDS
  __syncthreads();             // all waves' copies visible
  if (t < D_DIM) atomicAdd(&cvec, M_b[t] * y_s[(size_t)r * D_DIM + t]);
  __syncthreads();

  for (int l = wave; l < L_WORDS; l += 8) {
    float p = 0.f;
    for (int d = lane; d < D_DIM; d += 32) p += ew[l * D_DIM + d] * vv[d];
    p = wave_sum(p);
    if (lane == 0) dxs[l] = p + cvec;
  }
  __syncthreads();

  if (wave == 0) {
    float x = (lane < L_WORDS) ? dxs[lane] : -INFINITY;
    float m = wave_max(x);
    float e = __expf(x - m);
    float s = wave_sum(e);
    if (lane < L_WORDS) dxs[lane] = e / s;
  }
  __syncthreads();

  for (int d = t; d < D_DIM; d += 256) {
    float z = 0.f;
#pragma unroll 5
    for (int l = 0; l < L_WORDS; ++l) z += dxs[l] * ew[l * D_DIM + d];
    z_s[(size_t)r * D_DIM + d] = z;
  }
}

// ---------------------------------------------------------------------------
// Cosine hinge loss, one wave per row r.
// ---------------------------------------------------------------------------
__global__ void __launch_bounds__(256)
hinge_kernel(const float* __restrict__ r_s, const float* __restrict__ z_s,
             const float* __restrict__ z_n, float* __restrict__ out_abae,
             float* __restrict__ J_sum) {
  const int lane = threadIdx.x & 31;
  const int r = blockIdx.x * 8 + (threadIdx.x >> 5);
  if (r >= R_ROWS) return;
  const float* rp = r_s + (size_t)r * D_DIM;
  const float* zp = z_s + (size_t)r * D_DIM;

  float sr = 0.f, sz = 0.f, dzr = 0.f;
  for (int d = lane; d < D_DIM; d += 32) {
    float rv = rp[d], zv = zp[d];
    sr += rv * rv; sz += zv * zv; dzr += rv * zv;
  }
  sr = wave_sum(sr); sz = wave_sum(sz); dzr = wave_sum(dzr);
  const float nr = fmaxf(sqrtf(sr), 1e-12f);
  const float nz = fmaxf(sqrtf(sz), 1e-12f);
  const float c1 = dzr / (nr * nz);

  float part = 0.f;
  for (int j = 0; j < NUM_NEG; ++j) {
    const float* np = z_n + ((size_t)r * NUM_NEG + j) * D_DIM;
    float sn = 0.f, dn = 0.f;
    for (int d = lane; d < D_DIM; d += 32) {
      float nv = np[d];
      sn += nv * nv; dn += nv * rp[d];
    }
    sn = wave_sum(sn); dn = wave_sum(dn);
    const float c2 = dn / (fmaxf(sqrtf(sn), 1e-12f) * nr);
    const float loss = fmaxf(0.f, 1.f - (c1 - c2));
    if (lane == 0) { out_abae[r * NUM_NEG + j] = loss; part += loss; }
  }
  if (lane == 0) atomicAdd(J_sum, part);
}

// ---------------------------------------------------------------------------
// U_loss: column-normalized Gram of T_w (200x32), mean((G - I)^2). Single block.
// ---------------------------------------------------------------------------
__global__ void __launch_bounds__(256)
uloss_kernel(const float* __restrict__ T_w, float* __restrict__ U_out) {
  __shared__ float Ts[D_DIM * A_DIM];
  __shared__ float cn[A_DIM];
  __shared__ float accum;
  const int t = threadIdx.x;
  for (int i = t; i < D_DIM * A_DIM; i += 256) Ts[i] = T_w[i];
  if (t == 0) accum = 0.f;
  __syncthreads();
  if (t < A_DIM) {
    float s = 0.f;
    for (int d = 0; d < D_DIM; ++d) { float v = Ts[d * A_DIM + t]; s += v * v; }
    cn[t] = fmaxf(sqrtf(s), 1e-12f);
  }
  __syncthreads();
  float part = 0.f;
  for (int idx = t; idx < A_DIM * A_DIM; idx += 256) {
    const int a = idx >> 5, b = idx & 31;
    float g = 0.f;
    for (int d = 0; d < D_DIM; ++d) g += Ts[d * A_DIM + a] * Ts[d * A_DIM + b];
    g /= cn[a] * cn[b];
    const float dl = g - ((a == b) ? 1.f : 0.f);
    part += dl * dl;
  }
  atomicAdd(&accum, part);
  __syncthreads();
  if (t == 0) *U_out = accum / (float)(A_DIM * A_DIM);
}

// ---------------------------------------------------------------------------
// Sparse segment-sum scatter: user/item aspect pooling (fp32 global atomics).
// ---------------------------------------------------------------------------
__global__ void __launch_bounds__(256)
scatter_kernel(const int* __restrict__ uidx, const float* __restrict__ uval,
               const int* __restrict__ iidx, const float* __restrict__ ival,
               const float* __restrict__ p_t, float* __restrict__ out_user,
               float* __restrict__ out_item) {
  const int tid = blockIdx.x * 256 + threadIdx.x;
  if (tid >= 2 * NNZ_CNT) return;
  const int table = tid >= NNZ_CNT;
  const int n = tid - table * NNZ_CNT;
  const int* idx = table ? iidx : uidx;
  const float val = table ? ival[n] : uval[n];
  float* dst = table ? out_item : out_user;
  const int b = idx[n];
  const int r = idx[NNZ_CNT + n];
  const float* pr = p_t + (size_t)r * A_DIM;
  float* db = dst + (size_t)b * A_DIM;
#pragma unroll 8
  for (int a = 0; a < A_DIM; ++a) atomicAdd(&db[a], val * pr[a]);
}

// ---------------------------------------------------------------------------
// FM: per-b linear term + global scalar 0.5*sum(i1 - i2).
// ---------------------------------------------------------------------------
__global__ void __launch_bounds__(256)
fm_kernel(const float* __restrict__ out_user, const float* __restrict__ out_item,
          const float* __restrict__ fc_w, const float* __restrict__ fc_b,
          const float* __restrict__ fm_V, float* __restrict__ fm_lin,
          float* __restrict__ fm_s) {
  const int b = blockIdx.x * 256 + threadIdx.x;
  if (b >= B_BATCH) return;
  float oe[A_DIM];
  const float* ua = out_user + (size_t)b * A_DIM;
  const float* ia = out_item + (size_t)b * A_DIM;
#pragma unroll
  for (int a = 0; a < A_DIM; ++a) oe[a] = ua[a] * ia[a];
  float lin = fc_b[0];
#pragma unroll
  for (int a = 0; a < A_DIM; ++a) lin += oe[a] * fc_w[a];
  fm_lin[b] = lin;
  float part = 0.f;
  for (int k = 0; k < 10; ++k) {
    float s1 = 0.f, s2 = 0.f;
#pragma unroll
    for (int a = 0; a < A_DIM; ++a) {
      const float v = fm_V[a * 10 + k];
      s1 += oe[a] * v;
      s2 += oe[a] * oe[a] * v * v;
    }
    part += s1 * s1 - s2;
  }
  atomicAdd(fm_s, part);
}

__global__ void __launch_bounds__(256)
pred_kernel(const float* __restrict__ fm_lin, const float* __restrict__ fm_s,
            const float* __restrict__ label, float* __restrict__ out_pred,
            float* __restrict__ out_rloss, float* __restrict__ rating_sum) {
  const int b = blockIdx.x * 256 + threadIdx.x;
  if (b >= B_BATCH) return;
  const float pred = fm_lin[b] + 0.5f * fm_s[0] + 3.8f;
  out_pred[b] = pred;
  const float d = pred - label[b];
  const float rl = d * d;
  out_rloss[b] = rl;
  atomicAdd(rating_sum, rl);
}

__global__ void finalize_kernel(const float* __restrict__ scal, float* __restrict__ obj) {
  // scal: [0]=J_sum [1]=rating_sum [2]=fm_s [3]=U_loss
  *obj = scal[1] / (float)B_BATCH + scal[3] + scal[0] / (float)(R_ROWS * NUM_NEG);
}

__global__ void zero_kernel(float* __restrict__ p, int n) {
  const int i = blockIdx.x * 256 + threadIdx.x;
  if (i < n) p[i] = 0.f;
}

// ---------------------------------------------------------------------------
extern "C" void kernel_launch(void* const* d_in, const int* in_sizes, int n_in,
                              void* d_out, int out_size, void* d_ws, size_t ws_size,
                              hipStream_t stream) {
  const float* e_w   = (const float*)d_in[0];
  const float* y_s   = (const float*)d_in[1];
  const float* z_n   = (const float*)d_in[2];
  const float* label = (const float*)d_in[3];
  const float* uval  = (const float*)d_in[4];
  const float* ival  = (const float*)d_in[5];
  const float* M_w   = (const float*)d_in[6];
  const float* M_b   = (const float*)d_in[7];
  const float* W_w   = (const float*)d_in[8];
  const float* W_b   = (const float*)d_in[9];
  const float* T_w   = (const float*)d_in[10];
  const float* T_b   = (const float*)d_in[11];
  const float* fc_w  = (const float*)d_in[12];
  const float* fc_b  = (const float*)d_in[13];
  const float* fm_V  = (const float*)d_in[14];
  const int*   uidx  = (const int*)d_in[15];
  const int*   iidx  = (const int*)d_in[16];

  float* out = (float*)d_out;
  float* out_obj   = out;
  float* out_rloss = out + 1;
  float* out_abae  = out + 1 + B_BATCH;
  float* out_pred  = out + 1 + B_BATCH + R_ROWS * NUM_NEG;
  float* out_user  = out_pred + B_BATCH;
  float* out_item  = out_user + B_BATCH * A_DIM;

  float* ws    = (float*)d_ws;
  float* w_v   = ws;                                //  R*D
  float* w_zs  = w_v  + (size_t)R_ROWS * D_DIM;     //  R*D
  float* w_pt  = w_zs + (size_t)R_ROWS * D_DIM;     //  R*A
  float* w_rs  = w_pt + (size_t)R_ROWS * A_DIM;     //  R*D
  float* w_lin = w_rs + (size_t)R_ROWS * D_DIM;     //  B
  float* w_scal = w_lin + B_BATCH;                  //  4 scalars

  // Zero accumulators (deterministic: done every call).
  zero_kernel<<<(2 * B_BATCH * A_DIM + 255) / 256, 256, 0, stream>>>(
      out_user, 2 * B_BATCH * A_DIM);
  zero_kernel<<<1, 256, 0, stream>>>(w_scal, 4);

  const dim3 blk(256);
  const int mBlocks = (R_ROWS / 16 + 7) / 8;  // 157

  // v = y_s @ M_w                                  (WMMA f32 16x16x4)
  gemm_wmma_f32<<<dim3((D_DIM + 15) / 16, mBlocks), blk, 0, stream>>>(
      y_s, M_w, nullptr, w_v, R_ROWS, D_DIM, D_DIM, 0);
  // fused attention -> z_s (single async-staged streaming pass over e_w)
  attention_kernel<<<R_ROWS, blk, 0, stream>>>(e_w, y_s, M_b, w_v, w_zs);
  // p_t = z_s @ W_w^T + W_b                        (WMMA)
  gemm_wmma_f32<<<dim3((A_DIM + 15) / 16, mBlocks), blk, 0, stream>>>(
      w_zs, W_w, W_b, w_pt, R_ROWS, A_DIM, D_DIM, 1);
  // r_s = p_t @ T_w^T + T_b                        (WMMA)
  gemm_wmma_f32<<<dim3((D_DIM + 15) / 16, mBlocks), blk, 0, stream>>>(
      w_pt, T_w, T_b, w_rs, R_ROWS, D_DIM, A_DIM, 1);
  // hinge losses + J_sum
  hinge_kernel<<<R_ROWS / 8, blk, 0, stream>>>(w_rs, w_zs, z_n, out_abae, &w_scal[0]);
  // U_loss
  uloss_kernel<<<1, blk, 0, stream>>>(T_w, &w_scal[3]);
  // sparse aspect pooling
  scatter_kernel<<<(2 * NNZ_CNT + 255) / 256, blk, 0, stream>>>(
      uidx, uval, iidx, ival, w_pt, out_user, out_item);
  // FM
  fm_kernel<<<(B_BATCH + 255) / 256, blk, 0, stream>>>(
      out_user, out_item, fc_w, fc_b, fm_V, w_lin, &w_scal[2]);
  // prediction + rating loss
  pred_kernel<<<(B_BATCH + 255) / 256, blk, 0, stream>>>(
      w_lin, &w_scal[2], label, out_pred, out_rloss, &w_scal[1]);
  // obj
  finalize_kernel<<<1, 1, 0, stream>>>(w_scal, out_obj);

  (void)in_sizes; (void)n_in; (void)out_size; (void)ws_size;
}